// NCC_6090263625881
// MI455X (gfx1250) — compile-verified
//
#include <hip/hip_runtime.h>

typedef float v2f __attribute__((ext_vector_type(2)));
typedef float v8f __attribute__((ext_vector_type(8)));

namespace {
constexpr int VD = 160, VH = 192, VW = 224;       // volume (D,H,W)
constexpr int T1_STRIDE = 18;                      // padded w' stride (bank-conflict-free)
constexpr int T2_STRIDE = 19;                      // padded w' stride (bank-conflict-free)
// LDS layout (float offsets)
constexpr int I_OFF   = 0;                         // 24*24*24 = 13824
constexpr int J_OFF   = 13824;                     // 13824
constexpr int T1_OFF  = 27648;                     // 24*24 rows * 18 = 10368
constexpr int T2_OFF  = 38016;                     // 24*16 rows * 19 = 7296
constexpr int RED_OFF = 45312;                     // 8
constexpr int LDS_FLOATS = 45320;                  // 181,280 bytes (< 320KB/WGP)
}

// gfx1250 async global->LDS path (guarded; falls back to plain loads)
#if defined(__AMDGCN__) && defined(__has_builtin)
#  if __has_builtin(__builtin_amdgcn_global_load_async_to_lds_b32)
#    define HAS_ASYNC_LDS 1
#  endif
#endif
#ifndef HAS_ASYNC_LDS
#  define HAS_ASYNC_LDS 0
#endif

typedef __attribute__((address_space(1))) int* gint_t;   // global (AS1) int*
typedef __attribute__((address_space(3))) int* lint_t;   // LDS (AS3) int*

__device__ __forceinline__ void wait_async0() {
#if defined(__AMDGCN__)
#  if defined(__has_builtin) && __has_builtin(__builtin_amdgcn_s_wait_asynccnt)
  __builtin_amdgcn_s_wait_asynccnt(0);
#  else
  asm volatile("s_wait_asynccnt 0x0" ::: "memory");
#  endif
#endif
}

__device__ __forceinline__ float fieldval(int f, float i, float j) {
  switch (f) {
    case 0:  return i;
    case 1:  return j;
    case 2:  return i * i;
    case 3:  return j * j;
    default: return i * j;
  }
}

// Fused local-NCC: per 16^3 output tile, box-sum 5 fields via WMMA band-GEMMs.
extern "C" __global__ __launch_bounds__(256)
void ncc_main_kernel(const float* __restrict__ gI, const float* __restrict__ gJ,
                     float* __restrict__ partial)
{
  extern __shared__ float lds[];
  float* Il  = lds + I_OFF;
  float* Jl  = lds + J_OFF;
  float* t1  = lds + T1_OFF;
  float* t2  = lds + T2_OFF;
  float* red = lds + RED_OFF;

  const int tid   = threadIdx.x;
  const int wid   = __builtin_amdgcn_readfirstlane(tid >> 5); // scalar wave id 0..7
  const int lane  = tid & 31;
  const int n16   = lane & 15;    // N index / M index within fragments
  const int khalf = lane >> 4;    // 0: K rows {0,1}; 1: K rows {2,3}

  const int w0 = blockIdx.x * 16 - 4;
  const int h0 = blockIdx.y * 16 - 4;
  const int d0 = blockIdx.z * 16 - 4;

  // ---- load 24^3 halo tiles of I and J into LDS (zero padding outside) ----
  for (int e = tid; e < 13824; e += 256) {
    int d = e / 576;
    int r = e - d * 576;
    int h = r / 24;
    int w = r - h * 24;
    int gd = d0 + d, gh = h0 + h, gw = w0 + w;
    bool inb = (unsigned)gd < (unsigned)VD && (unsigned)gh < (unsigned)VH &&
               (unsigned)gw < (unsigned)VW;
    long idx = ((long)gd * VH + gh) * VW + gw;
#if HAS_ASYNC_LDS
    if (inb) {
      __builtin_amdgcn_global_load_async_to_lds_b32(
          (gint_t)(gI + idx), (lint_t)(Il + e), 0, 0);
      __builtin_amdgcn_global_load_async_to_lds_b32(
          (gint_t)(gJ + idx), (lint_t)(Jl + e), 0, 0);
    } else {
      Il[e] = 0.0f;
      Jl[e] = 0.0f;
    }
#else
    float vi = 0.0f, vj = 0.0f;
    if (inb) { vi = gI[idx]; vj = gJ[idx]; }
    Il[e] = vi;
    Jl[e] = vj;
#endif
  }
#if HAS_ASYNC_LDS
  wait_async0();
#endif

  // ---- constant band-of-ones B fragments: B[k][n] = (n <= k <= n+8) ----
  // B 4x16 f32 layout (mirrors A 16x4): v0/v1 hold K rows (2*khalf)/(2*khalf+1).
  v2f bfrag[6];
#pragma unroll
  for (int ks = 0; ks < 6; ++ks) {
    int r0 = ks * 4 + 2 * khalf;
    bfrag[ks].x = (r0     >= n16 && r0     <= n16 + 8) ? 1.0f : 0.0f;
    bfrag[ks].y = (r0 + 1 >= n16 && r0 + 1 <= n16 + 8) ? 1.0f : 0.0f;
  }

  float s[5][2][8];   // [field][d-pass tile][C vgpr] box sums, kept in registers

  __syncthreads();

#pragma unroll
  for (int f = 0; f < 5; ++f) {
    // ---- W pass: lines (d,h) in 24x24, out 16 w. 36 M-tiles of 16 lines ----
    for (int t = wid; t < 36; t += 8) {
      int L = t * 16 + n16;                 // line id = d*24+h, A fragment M row
      const float* ip = Il + L * 24;
      const float* jp = Jl + L * 24;
      v8f c = {};
#pragma unroll
      for (int ks = 0; ks < 6; ++ks) {
        int k = ks * 4 + 2 * khalf;
        v2f a;
        a.x = fieldval(f, ip[k],     jp[k]);
        a.y = fieldval(f, ip[k + 1], jp[k + 1]);
        c = __builtin_amdgcn_wmma_f32_16x16x4_f32(false, a, false, bfrag[ks],
                                                  (short)0, c, false, false);
      }
#pragma unroll
      for (int r = 0; r < 8; ++r) {
        int Lm = t * 16 + r + 8 * khalf;    // C row M -> line id (= d*24+h)
        t1[Lm * T1_STRIDE + n16] = c[r];    // tmp1[line][w']
      }
    }
    __syncthreads();

    // ---- H pass: per d (24), lines = 16 w', K = 24 h, out 16 h' ----
    for (int d = wid; d < 24; d += 8) {
      v8f c = {};
#pragma unroll
      for (int ks = 0; ks < 6; ++ks) {
        int k = ks * 4 + 2 * khalf;
        v2f a;
        a.x = t1[(d * 24 + k    ) * T1_STRIDE + n16];
        a.y = t1[(d * 24 + k + 1) * T1_STRIDE + n16];
        c = __builtin_amdgcn_wmma_f32_16x16x4_f32(false, a, false, bfrag[ks],
                                                  (short)0, c, false, false);
      }
#pragma unroll
      for (int r = 0; r < 8; ++r) {
        // out[m=w'][n=h'] -> tmp2[d][h'][w']
        t2[(d * 16 + n16) * T2_STRIDE + (r + 8 * khalf)] = c[r];
      }
    }
    __syncthreads();

    // ---- D pass: tile t = h' (wave gets 2), lines = 16 w', K = 24 d ----
#pragma unroll
    for (int t2i = 0; t2i < 2; ++t2i) {
      int t = wid * 2 + t2i;                // h'
      v8f c = {};
#pragma unroll
      for (int ks = 0; ks < 6; ++ks) {
        int k = ks * 4 + 2 * khalf;
        v2f a;
        a.x = t2[((k    ) * 16 + t) * T2_STRIDE + n16];
        a.y = t2[((k + 1) * 16 + t) * T2_STRIDE + n16];
        c = __builtin_amdgcn_wmma_f32_16x16x4_f32(false, a, false, bfrag[ks],
                                                  (short)0, c, false, false);
      }
#pragma unroll
      for (int r = 0; r < 8; ++r) s[f][t2i][r] = c[r];
    }
    __syncthreads();
  }

  // ---- pointwise NCC (literal reference math) + block reduction ----
  float acc = 0.0f;
#pragma unroll
  for (int t2i = 0; t2i < 2; ++t2i) {
#pragma unroll
    for (int r = 0; r < 8; ++r) {
      float sI  = s[0][t2i][r], sJ  = s[1][t2i][r];
      float sI2 = s[2][t2i][r], sJ2 = s[3][t2i][r], sIJ = s[4][t2i][r];
      float uI = sI / 729.0f, uJ = sJ / 729.0f;
      float cross = sIJ - uJ * sI - uI * sJ + uI * uJ * 729.0f;
      float Ivar  = sI2 - 2.0f * uI * sI + uI * uI * 729.0f;
      float Jvar  = sJ2 - 2.0f * uJ * sJ + uJ * uJ * 729.0f;
      acc += cross * cross / (Ivar * Jvar + 1e-5f);
    }
  }
  for (int o = 16; o > 0; o >>= 1) acc += __shfl_xor(acc, o, 32);
  if (lane == 0) red[wid] = acc;
  __syncthreads();
  if (tid == 0) {
    float b = 0.0f;
#pragma unroll
    for (int i = 0; i < 8; ++i) b += red[i];
    partial[(blockIdx.z * gridDim.y + blockIdx.y) * gridDim.x + blockIdx.x] = b;
  }
}

extern "C" __global__ __launch_bounds__(256)
void ncc_finalize_kernel(const float* __restrict__ partial, float* __restrict__ out,
                         int n)
{
  __shared__ float red[8];
  float a = 0.0f;
  for (int i = threadIdx.x; i < n; i += 256) a += partial[i];
  for (int o = 16; o > 0; o >>= 1) a += __shfl_xor(a, o, 32);
  int wid = threadIdx.x >> 5, lane = threadIdx.x & 31;
  if (lane == 0) red[wid] = a;
  __syncthreads();
  if (threadIdx.x == 0) {
    float t = 0.0f;
#pragma unroll
    for (int i = 0; i < 8; ++i) t += red[i];
    out[0] = -t / (float)((long)VD * VH * VW);
  }
}

extern "C" void kernel_launch(void* const* d_in, const int* in_sizes, int n_in,
                              void* d_out, int out_size, void* d_ws, size_t ws_size,
                              hipStream_t stream) {
  (void)in_sizes; (void)n_in; (void)out_size; (void)ws_size;
  const float* I = (const float*)d_in[0];
  const float* J = (const float*)d_in[1];
  float* partial = (float*)d_ws;
  float* out = (float*)d_out;

  dim3 grid(VW / 16, VH / 16, VD / 16);   // 14 x 12 x 10 = 1680 blocks
  const int nPartial = (VW / 16) * (VH / 16) * (VD / 16);
  const size_t ldsBytes = (size_t)LDS_FLOATS * sizeof(float);

  // Allow >64KB dynamic LDS (MI455X WGP has 320KB).
  (void)hipFuncSetAttribute((const void*)ncc_main_kernel,
                            hipFuncAttributeMaxDynamicSharedMemorySize,
                            (int)ldsBytes);

  ncc_main_kernel<<<grid, 256, ldsBytes, stream>>>(I, J, partial);
  ncc_finalize_kernel<<<1, 256, 0, stream>>>(partial, out, nPartial);
}